// ClusterLookup_50491635532314
// MI455X (gfx1250) — compile-verified
//
#include <hip/hip_runtime.h>
#include <hip/hip_bf16.h>

// ---------------------------------------------------------------------------
// ClusterLookup fused kernel for MI455X (gfx1250, wave32, WMMA f32 16x16x4)
//   P = B*H*W = 147456 pixels, C = 512, N = 27 clusters (padded to 32)
//   GEMM [P x C] x [C x 32] fused with per-pixel L2 norm, softmax, loss.
// ---------------------------------------------------------------------------

typedef __attribute__((ext_vector_type(2))) float v2f;
typedef __attribute__((ext_vector_type(8))) float v8f;

#define CDIM   512
#define NCLUS  27
#define HW     9216            // 96*96
#define PIXTOT 147456          // 16*96*96
#define NTILES 9216            // PIXTOT / 16
#define WAVES_PER_BLOCK 8
#define NBLOCKS (NTILES / WAVES_PER_BLOCK)   // 1152
#define KCHUNKS (CDIM / 4)     // 128 wmma steps of K=4

// ---------------------------------------------------------------------------
// Kernel 1: L2-normalize clusters [27 x 512] -> ws.cnorm
// ---------------------------------------------------------------------------
__global__ __launch_bounds__(256) void norm_clusters_kernel(
    const float* __restrict__ clusters, float* __restrict__ cnorm)
{
    const int n = blockIdx.x;              // 0..26
    const int t = threadIdx.x;             // 0..255
    __shared__ float red[8];

    float a = clusters[n * CDIM + t];
    float b = clusters[n * CDIM + 256 + t];
    float s = a * a + b * b;
    #pragma unroll
    for (int d = 1; d < 32; d <<= 1) s += __shfl_xor(s, d, 32);
    if ((t & 31) == 0) red[t >> 5] = s;
    __syncthreads();
    float tot = 0.0f;
    #pragma unroll
    for (int i = 0; i < 8; ++i) tot += red[i];
    const float inv = 1.0f / fmaxf(sqrtf(tot), 1e-12f);
    cnorm[n * CDIM + t]       = a * inv;
    cnorm[n * CDIM + 256 + t] = b * inv;
}

// ---------------------------------------------------------------------------
// Kernel 2: fused GEMM(WMMA f32) + pixel-norm + softmax + probs + loss part.
// One wave = one 16-pixel tile. Block = 8 waves.
// LDS holds clusters pre-swizzled into WMMA B-operand layout:
//   blds[((nt*128 + c)*32 + lane)*2 + t]  (2 N-tiles x 128 K-chunks) = 64 KB
// ---------------------------------------------------------------------------
__global__ __launch_bounds__(256) void cluster_lookup_main(
    const float* __restrict__ x,        // [16,512,96,96]
    const float* __restrict__ cnorm,    // [27,512] normalized
    const float* __restrict__ alphap,   // [1]
    float* __restrict__ probs,          // [16,27,96,96]
    float* __restrict__ partials)       // [NTILES]
{
    __shared__ float blds[2 * KCHUNKS * 64];   // 16384 floats = 64 KB

    const int t = threadIdx.x;
    // Cooperative fill of B in WMMA lane layout (cols 27..31 zero-padded).
    for (int idx = t; idx < 2 * KCHUNKS * 64; idx += 256) {
        const int nt   = idx >> 13;        // N-tile (0/1)
        const int r    = idx & 8191;
        const int c    = r >> 6;           // K-chunk
        const int q    = r & 63;
        const int lane = q >> 1;
        const int tt   = q & 1;
        const int k    = 4 * c + 2 * (lane >> 4) + tt;
        const int n    = nt * 16 + (lane & 15);
        blds[idx] = (n < NCLUS) ? cnorm[n * CDIM + k] : 0.0f;
    }
    __syncthreads();

    const int lane = t & 31;
    const int wave = t >> 5;
    const int tile = blockIdx.x * WAVES_PER_BLOCK + wave;   // exact: 9216 tiles

    const int pixGlobal = tile * 16;
    const int b    = pixGlobal / HW;
    const int pix0 = pixGlobal - b * HW;

    // A operand: lane L -> pixel M = L&15 ; K offset 2*(L>>4) within chunk.
    const float* __restrict__ xb = x + b * (CDIM * HW) + pix0 + (lane & 15);
    const int koff = (lane >> 4) * 2;

    const float alpha = alphap[0];

    v8f acc0 = {};   // cols  0..15
    v8f acc1 = {};   // cols 16..31 (27..31 padded w/ zero B)
    float ssq = 0.0f;

    const float* __restrict__ b0 = &blds[lane * 2];
    const float* __restrict__ b1 = &blds[8192 + lane * 2];

    #pragma unroll 4
    for (int c = 0; c < KCHUNKS; ++c) {
        const int k = 4 * c + koff;
        const float a0 = xb[k * HW];
        const float a1 = xb[(k + 1) * HW];
        ssq = fmaf(a0, a0, ssq);
        ssq = fmaf(a1, a1, ssq);
        v2f A; A.x = a0; A.y = a1;
        const v2f B0 = *(const v2f*)(b0 + c * 64);
        const v2f B1 = *(const v2f*)(b1 + c * 64);
        acc0 = __builtin_amdgcn_wmma_f32_16x16x4_f32(
            false, A, false, B0, (short)0, acc0, false, false);
        acc1 = __builtin_amdgcn_wmma_f32_16x16x4_f32(
            false, A, false, B1, (short)0, acc1, false, false);
    }

    // Per-pixel sum of squares: lanes L and L^16 each hold half the K range.
    const float ssqFull = ssq + __shfl_xor(ssq, 16, 32);     // pixel = lane&15
    const float scale = 1.0f / fmaxf(sqrtf(ssqFull), 1e-12f);

    // C/D layout: VGPR j, lanes 0-15 -> row M=j, col=lane ;
    //                      lanes 16-31 -> row M=j+8, col=lane-16.
    const int half8 = (lane >> 4) << 3;
    const int col0  = lane & 15;
    const int col1  = 16 + col0;
    const bool v1   = (col1 < NCLUS);

    float lossAcc = 0.0f;

    #pragma unroll
    for (int j = 0; j < 8; ++j) {
        const float rs  = __shfl(scale, j + half8, 32);      // row's 1/||x||
        const float in0 = acc0[j] * rs;
        const float in1 = acc1[j] * rs;
        const float t0  = alpha * in0;
        const float t1  = v1 ? (alpha * in1) : -3.0e38f;

        float m = fmaxf(t0, t1);
        m = fmaxf(m, __shfl_xor(m, 1, 32));
        m = fmaxf(m, __shfl_xor(m, 2, 32));
        m = fmaxf(m, __shfl_xor(m, 4, 32));
        m = fmaxf(m, __shfl_xor(m, 8, 32));

        const float e0 = __expf(t0 - m);
        const float e1 = v1 ? __expf(t1 - m) : 0.0f;

        float s = e0 + e1;
        s += __shfl_xor(s, 1, 32);
        s += __shfl_xor(s, 2, 32);
        s += __shfl_xor(s, 4, 32);
        s += __shfl_xor(s, 8, 32);

        float d = e0 * in0 + e1 * in1;
        d += __shfl_xor(d, 1, 32);
        d += __shfl_xor(d, 2, 32);
        d += __shfl_xor(d, 4, 32);
        d += __shfl_xor(d, 8, 32);

        const float invs = 1.0f / s;
        const int pixRow = pix0 + j + half8;

        probs[(b * NCLUS + col0) * HW + pixRow] = e0 * invs;
        if (v1)
            probs[(b * NCLUS + col1) * HW + pixRow] = e1 * invs;
        if (col0 == 0)
            lossAcc += d * invs;     // lanes 0 & 16 each cover 8 rows
    }

    const float tileLoss = lossAcc + __shfl_xor(lossAcc, 16, 32);
    if (lane == 0) partials[tile] = tileLoss;
}

// ---------------------------------------------------------------------------
// Kernel 3: deterministic loss reduction (fixed order, no float atomics)
// ---------------------------------------------------------------------------
__global__ __launch_bounds__(256) void reduce_loss_kernel(
    const float* __restrict__ partials, float* __restrict__ out)
{
    __shared__ float sm[256];
    float s = 0.0f;
    for (int i = threadIdx.x; i < NTILES; i += 256) s += partials[i];
    sm[threadIdx.x] = s;
    __syncthreads();
    for (int off = 128; off > 0; off >>= 1) {
        if (threadIdx.x < off) sm[threadIdx.x] += sm[threadIdx.x + off];
        __syncthreads();
    }
    if (threadIdx.x == 0) out[0] = -sm[0] / (float)PIXTOT;
}

// ---------------------------------------------------------------------------
extern "C" void kernel_launch(void* const* d_in, const int* in_sizes, int n_in,
                              void* d_out, int out_size, void* d_ws, size_t ws_size,
                              hipStream_t stream) {
    const float* x        = (const float*)d_in[0];   // [16,512,96,96]
    const float* clusters = (const float*)d_in[1];   // [27,512]
    const float* alpha    = (const float*)d_in[2];   // [1]

    float* out      = (float*)d_out;                 // [0]=loss, [1..]=probs
    float* cnorm    = (float*)d_ws;                  // 27*512 floats
    float* partials = cnorm + NCLUS * CDIM;          // NTILES floats

    norm_clusters_kernel<<<NCLUS, 256, 0, stream>>>(clusters, cnorm);
    cluster_lookup_main<<<NBLOCKS, 256, 0, stream>>>(x, cnorm, alpha,
                                                     out + 1, partials);
    reduce_loss_kernel<<<1, 256, 0, stream>>>(partials, out);
}